// QwenLinearAttentionBlock_77773267796312
// MI455X (gfx1250) — compile-verified
//
#include <hip/hip_runtime.h>
#include <math.h>
#include <stdint.h>

// Problem constants (from reference)
#define HSZ 1024   // hidden size
#define NH  16     // heads
#define HD  128    // head dim
#define ID  2048   // inner dim = NH*HD
// workspace layout (floats): [0,6144) qkv | [6144,8192) z | [8192,8208) a-logits | [8208,10256) gated y
#define OFF_Q    0
#define OFF_K    ID
#define OFF_V    (2*ID)
#define OFF_Z    (3*ID)
#define OFF_A    (3*ID + ID)
#define OFF_YG   (3*ID + ID + NH)

typedef float v2f __attribute__((ext_vector_type(2)));
typedef float v8f __attribute__((ext_vector_type(8)));

// LDS row stride for a 16x64 weight panel: 64 + 4 pad
//  - (68*r + 4*c)*4 bytes is 16B aligned -> b128 LDS writes OK
//  - bank(lane n, col k) = (4n + k) % 64 -> 16 distinct banks, conflict-free
#define WROW   68
#define PANEL  (16 * WROW)

// ---------------- CDNA5 async global->LDS path (probe-guarded) ----------------
#if defined(__has_builtin)
#if __has_builtin(__builtin_amdgcn_global_load_async_to_lds_b128)
#define USE_ASYNC_LDS 1
#endif
#endif

#if __has_builtin(__builtin_amdgcn_s_wait_asynccnt)
#define WAIT_ASYNCCNT(n) __builtin_amdgcn_s_wait_asynccnt(n)
#else
#define WAIT_ASYNCCNT(n) asm volatile("s_wait_asynccnt " #n ::: "memory")
#endif

#if __has_builtin(__builtin_amdgcn_s_wait_dscnt)
#define WAIT_DSCNT(n) __builtin_amdgcn_s_wait_dscnt(n)
#else
#define WAIT_DSCNT(n) asm volatile("s_wait_dscnt " #n ::: "memory")
#endif

#ifdef USE_ASYNC_LDS
// Builtin prototype (from the round-2 diagnostic): the pointer params are
// pointers to 16-byte int vectors: "vV4i*1 V4i*3 Ii Ii".
typedef int v4i_b128 __attribute__((vector_size(16)));
typedef __attribute__((address_space(1))) v4i_b128* gas_b128_p;  // global src
typedef __attribute__((address_space(3))) v4i_b128* las_b128_p;  // LDS dst

// Stage one 16x64 f32 panel (16 rows of W, row length K, starting at col kb)
// into an LDS panel via 8 GLOBAL_LOAD_ASYNC_TO_LDS_B128 (ASYNCcnt += 8).
// Flat->LDS offset: low 32 bits of the flat shared address are the LDS byte
// offset (aperture truncation rule); flat == global for global memory.
__device__ __forceinline__ void stage_panel_async(const float* __restrict__ Wrows,
                                                  int K, int kb, float* buf, int lane)
{
#pragma unroll
    for (int i = 0; i < 8; ++i) {
        const int idx = lane + 32 * i;        // 0..255 chunks of float4
        const int r   = idx >> 4;             // row 0..15
        const int c4  = (idx & 15) << 2;      // col 0,4,...,60
        const float* g = Wrows + (size_t)r * K + kb + c4;
        float*       l = buf + r * WROW + c4;
        __builtin_amdgcn_global_load_async_to_lds_b128(
            (gas_b128_p)(uintptr_t)g, (las_b128_p)(uintptr_t)l,
            /*offset=*/0, /*cpol=*/0);
    }
}
#endif

// GEMV of x[K] against 16 consecutive rows of W (row-major, row length K),
// using V_WMMA_F32_16X16X4_F32 with x broadcast across all 16 A-rows.
// A 16x4 f32 layout (ISA 7.12.2): V0 = {lanes0-15:K0, lanes16-31:K2},
//                                 V1 = {lanes0-15:K1, lanes16-31:K3}.
// B 4x16 f32 built as the mirrored layout (lane = N, halves/VGPRs = K).
// D layout: c[0] lanes 0-15 hold M=0, N=lane -> result per output n.
// `lds` must hold 2*PANEL floats (double buffer).
__device__ __forceinline__ v8f wmma_gemv16_tile(const float* __restrict__ Wrows,
                                                const float* __restrict__ xv,
                                                int K, float* lds)
{
    const int lane = threadIdx.x & 31;
    const int n    = lane & 15;
    const int half = lane >> 4;
    v8f c = {0.f, 0.f, 0.f, 0.f, 0.f, 0.f, 0.f, 0.f};

#ifdef USE_ASYNC_LDS
    // Double-buffered async pipeline: ASYNCcnt-tracked copies straight to LDS,
    // overlapped with the WMMA chain on the resident panel.
    const int nch = K >> 6;                      // panels of 64 K-values
    stage_panel_async(Wrows, K, 0, lds, lane);   // prime buffer 0 (+8 async)
    for (int ch = 0; ch < nch; ++ch) {
        float* cur = lds + (ch & 1) * PANEL;
        if (ch + 1 < nch) {
            // Guard DS-read vs async-LDS-write race on the buffer being refilled
            WAIT_DSCNT(0);
            stage_panel_async(Wrows, K, (ch + 1) << 6,
                              lds + ((ch + 1) & 1) * PANEL, lane);  // +8 async
            // Async loads complete in order: <=8 outstanding means the current
            // panel's 8 copies have landed while the next 8 are in flight.
            WAIT_ASYNCCNT(8);
        } else {
            WAIT_ASYNCCNT(0);
        }
        asm volatile("" ::: "memory");           // keep LDS reads below the wait
        const int kb = ch << 6;
#pragma unroll
        for (int kl = 0; kl < 64; kl += 4) {
            const float2 xa = *reinterpret_cast<const float2*>(xv + kb + kl + 2 * half);
            const float2 wb = *reinterpret_cast<const float2*>(cur + n * WROW + kl + 2 * half);
            v2f a; a[0] = xa.x; a[1] = xa.y;
            v2f b; b[0] = wb.x; b[1] = wb.y;
            c = __builtin_amdgcn_wmma_f32_16x16x4_f32(
                    false, a, false, b, (short)0, c, false, false);
        }
    }
#else
    // Fallback: synchronous staging
    for (int kb = 0; kb < K; kb += 64) {
#pragma unroll
        for (int i = 0; i < 8; ++i) {
            const int idx = lane + 32 * i;
            const int r   = idx >> 4;
            const int c4  = (idx & 15) << 2;
            *reinterpret_cast<float4*>(lds + r * WROW + c4) =
                *reinterpret_cast<const float4*>(Wrows + (size_t)r * K + kb + c4);
        }
        __syncthreads();
#pragma unroll
        for (int kl = 0; kl < 64; kl += 4) {
            const float2 xa = *reinterpret_cast<const float2*>(xv + kb + kl + 2 * half);
            const float2 wb = *reinterpret_cast<const float2*>(lds + n * WROW + kl + 2 * half);
            v2f a; a[0] = xa.x; a[1] = xa.y;
            v2f b; b[0] = wb.x; b[1] = wb.y;
            c = __builtin_amdgcn_wmma_f32_16x16x4_f32(
                    false, a, false, b, (short)0, c, false, false);
        }
        __syncthreads();
    }
#endif
    return c;
}

// ---- Kernel 1: fused projections (qkv | z | a-logits), 513 x 16-output WMMA tiles
__global__ __launch_bounds__(32)
void proj_kernel(const float* __restrict__ x,
                 const float* __restrict__ Wqkv,
                 const float* __restrict__ Wz,
                 const float* __restrict__ Wa,
                 float* __restrict__ p)
{
    __shared__ float lds[2 * PANEL];
    const int tile = blockIdx.x;
    const float* Wrows;
    int off;
    if (tile < 384) {                         // W_qkv: 6144 rows
        Wrows = Wqkv + (size_t)tile * 16 * HSZ;
        off   = tile * 16;
    } else if (tile < 512) {                  // W_z: 2048 rows
        Wrows = Wz + (size_t)(tile - 384) * 16 * HSZ;
        off   = OFF_Z + (tile - 384) * 16;
    } else {                                  // W_a: exactly 16 rows
        Wrows = Wa;
        off   = OFF_A;
    }
    v8f c = wmma_gemv16_tile(Wrows, x, HSZ, lds);
    const int lane = threadIdx.x & 31;
    if (lane < 16) p[off + lane] = c[0];
}

// ---- Kernel 2: per-head state update + q readout + RMSNorm + SiLU gate.
// One block per head; thread e owns state column e (coalesced streaming).
__global__ __launch_bounds__(HD)
void state_kernel(const float* __restrict__ rnn_state,
                  const float* __restrict__ p,
                  const float* __restrict__ norm_w,
                  float* __restrict__ new_state_out,
                  float* __restrict__ ygated)
{
    __shared__ float red[HD];
    const int h = blockIdx.x;
    const int e = threadIdx.x;

    const float alogit = p[OFF_A + h];
    const float decay  = 1.f / (1.f + __expf(-alogit));

    const float* q  = p + OFF_Q + h * HD;
    const float* k  = p + OFF_K + h * HD;
    const float  ve = p[OFF_V + h * HD + e];

    const float* rs = rnn_state     + (size_t)h * HD * HD;
    float*       ns = new_state_out + (size_t)h * HD * HD;

    float acc = 0.f;
#pragma unroll 4
    for (int d = 0; d < HD; ++d) {
        const float s = rs[d * HD + e] * decay + k[d] * ve;  // new_state[h][d][e]
        ns[d * HD + e] = s;
        acc += q[d] * s;                                     // y[h][e]
    }

    // RMSNorm over head dim
    red[e] = acc * acc;
    __syncthreads();
    for (int s = HD / 2; s > 0; s >>= 1) {
        if (e < s) red[e] += red[e + s];
        __syncthreads();
    }
    const float inv = rsqrtf(red[0] / (float)HD + 1e-6f);
    const float yn  = acc * inv * norm_w[e];

    // SiLU gate: y * (z * sigmoid(z))
    const float z   = p[OFF_Z + h * HD + e];
    const float sig = 1.f / (1.f + __expf(-z));
    ygated[h * HD + e] = yn * (z * sig);
}

// ---- Kernel 3: output projection, 64 x 16-output WMMA tiles against W_out[1024,2048]
__global__ __launch_bounds__(32)
void out_kernel(const float* __restrict__ ygated,
                const float* __restrict__ Wout,
                float* __restrict__ out)
{
    __shared__ float lds[2 * PANEL];
    const int tile = blockIdx.x;
    v8f c = wmma_gemv16_tile(Wout + (size_t)tile * 16 * ID, ygated, ID, lds);
    const int lane = threadIdx.x & 31;
    if (lane < 16) out[tile * 16 + lane] = c[0];
}

extern "C" void kernel_launch(void* const* d_in, const int* in_sizes, int n_in,
                              void* d_out, int out_size, void* d_ws, size_t ws_size,
                              hipStream_t stream) {
    (void)in_sizes; (void)n_in; (void)out_size; (void)ws_size;

    const float* x          = (const float*)d_in[0];
    const float* rnn_state  = (const float*)d_in[1];
    const float* conv_state = (const float*)d_in[2];
    const float* W_qkv      = (const float*)d_in[3];
    const float* W_z        = (const float*)d_in[4];
    const float* W_a        = (const float*)d_in[5];
    const float* norm_w     = (const float*)d_in[6];
    const float* W_out      = (const float*)d_in[7];

    float* out      = (float*)d_out;                 // [1024]
    float* ns_out   = out + HSZ;                     // [16*128*128]
    float* conv_out = ns_out + (size_t)NH * HD * HD; // [3*ID*3]

    float* ws     = (float*)d_ws;
    float* p      = ws;            // 8208 floats: qkv | z | a
    float* ygated = ws + OFF_YG;   // 2048 floats

    // 1) All projections (qkv, z, decay logits): 513 WMMA tiles
    proj_kernel<<<dim3(384 + 128 + 1), dim3(32), 0, stream>>>(x, W_qkv, W_z, W_a, p);

    // 2) State update + readout + RMSNorm + gate (writes new_state to d_out)
    state_kernel<<<dim3(NH), dim3(HD), 0, stream>>>(rnn_state, p, norm_w, ns_out, ygated);

    // 3) Output projection: 64 WMMA tiles
    out_kernel<<<dim3(HSZ / 16), dim3(32), 0, stream>>>(ygated, W_out, out);

    // 4) conv_state pass-through
    (void)hipMemcpyAsync(conv_out, conv_state, (size_t)3 * ID * 3 * sizeof(float),
                         hipMemcpyDeviceToDevice, stream);
}